// FeatureVolumeManager_84731114815582
// MI455X (gfx1250) — compile-verified
//
#include <hip/hip_runtime.h>
#include <hip/hip_bf16.h>
#include <math.h>

typedef __attribute__((ext_vector_type(2))) float v2f;
typedef __attribute__((ext_vector_type(8))) float v8f;

#define S_VIEWS   7
#define C_CH      16
#define H_DIM     64
#define W_DIM     96
#define D_BINS    64
#define N_PIX     (H_DIM * W_DIM)          // 6144
#define PT_PER_ROW (W_DIM / 16)            // 6
#define PIX_TILES (N_PIX / 16)             // 384
#define DEPTH_TILES (D_BINS / 16)          // 4

// Plane-sweep cost volume. One wave32 per (16-pixel tile, 16-depth tile).
// Channel contraction done with V_WMMA_F32_16X16X4_F32 (diagonal trick):
//   A[m][c] = masked bilinear warped feature (pixel m, channel c) for (s, d)
//   B[c][n] = cur_feats channel c at pixel n
//   D[m][n] = <warped(m), cur(n)>  -> diagonal m==n is the wanted dot.
// Accumulate over the 7 views in the f32 WMMA accumulator (mask folded into
// the bilinear weights), then divide the diagonal by (mask_sum + 1e-4).
__global__ __launch_bounds__(32)
void FeatureVolume_cost_kernel(const float* __restrict__ cur_feats,
                               const float* __restrict__ src_feats,
                               const float* __restrict__ src_ext,
                               const float* __restrict__ src_Ks,
                               const float* __restrict__ cur_invK,
                               const float* __restrict__ min_depth,
                               const float* __restrict__ max_depth,
                               float* __restrict__ out)
{
    __shared__ float Psh[S_VIEWS][12];   // rows 0..2 of (K @ ext), 4 cols each

    const int lane = threadIdx.x;        // 0..31
    const int blk  = blockIdx.x;         // 0..PIX_TILES*DEPTH_TILES-1
    const int pt   = blk % PIX_TILES;
    const int dt   = blk / PIX_TILES;

    // ---- P = (K @ ext)[:3,:4] per view, computed once per block into LDS ----
    for (int idx = lane; idx < S_VIEWS * 12; idx += 32) {
        const int s   = idx / 12;
        const int r   = (idx % 12) / 4;
        const int col = idx % 4;
        const float* K = src_Ks  + s * 16;
        const float* E = src_ext + s * 16;
        float acc = 0.0f;
        #pragma unroll
        for (int k = 0; k < 4; ++k) acc = fmaf(K[r * 4 + k], E[k * 4 + col], acc);
        Psh[s][r * 4 + col] = acc;
    }
    __syncthreads();

    const int row  = pt / PT_PER_ROW;            // 0..63
    const int colb = (pt % PT_PER_ROW) * 16;     // 0,16,...,80
    const int m    = lane & 15;                  // pixel within tile (A row / B col)
    const int h    = lane >> 4;                  // half-wave selector

    const float gx = (float)(colb + m) + 0.5f;
    const float gy = (float)row + 0.5f;

    // ray = invK[:3,:3] @ (gx, gy, 1)
    const float r0 = fmaf(cur_invK[0], gx, fmaf(cur_invK[1], gy, cur_invK[2]));
    const float r1 = fmaf(cur_invK[4], gx, fmaf(cur_invK[5], gy, cur_invK[6]));
    const float r2 = fmaf(cur_invK[8], gx, fmaf(cur_invK[9], gy, cur_invK[10]));

    // ---- B matrix (cur_feats at the 16 tile pixels), 4 K-steps of K=4 ----
    // lane L, step j, VGPR v  ->  channel c = 4j + 2h + v, column n = L&15
    const int pix = row * W_DIM + colb + m;
    v2f bmat[4];
    #pragma unroll
    for (int j = 0; j < 4; ++j) {
        const int c0 = 4 * j + 2 * h;
        bmat[j].x = cur_feats[c0 * N_PIX + pix];
        bmat[j].y = cur_feats[(c0 + 1) * N_PIX + pix];
    }

    const float dmin = min_depth[0];
    const float dmax = max_depth[0];
    const float lrat = logf(dmax / dmin);

    for (int dl = 0; dl < 16; ++dl) {
        const int   dg    = dt * 16 + dl;
        const float depth = dmin * expf(lrat * ((float)dg * (1.0f / (float)(D_BINS - 1))));

        v8f cacc = {0.f, 0.f, 0.f, 0.f, 0.f, 0.f, 0.f, 0.f};
        float msum = 0.0f;

        for (int s = 0; s < S_VIEWS; ++s) {
            const float* P = &Psh[s][0];
            // cam = depth * (P[:, :3] @ ray) + P[:, 3]
            const float cx = fmaf(depth, fmaf(P[0], r0, fmaf(P[1], r1, P[2]  * r2)), P[3]);
            const float cy = fmaf(depth, fmaf(P[4], r0, fmaf(P[5], r1, P[6]  * r2)), P[7]);
            const float cz = fmaf(depth, fmaf(P[8], r0, fmaf(P[9], r1, P[10] * r2)), P[11]);

            const float mask = (cz > 0.0f) ? 1.0f : 0.0f;
            msum += mask;

            const float iz = 1.0f / (cz + 1e-8f);
            const float x  = cx * iz - 0.5f;
            const float y  = cy * iz - 0.5f;
            const float x0f = floorf(x), y0f = floorf(y);
            const float x1f = x0f + 1.0f, y1f = y0f + 1.0f;
            const float wx1 = x - x0f, wy1 = y - y0f;
            const float wx0 = 1.0f - wx1, wy0 = 1.0f - wy1;

            const float vx0 = (x0f >= 0.0f && x0f <= (float)(W_DIM - 1)) ? 1.0f : 0.0f;
            const float vx1 = (x1f >= 0.0f && x1f <= (float)(W_DIM - 1)) ? 1.0f : 0.0f;
            const float vy0 = (y0f >= 0.0f && y0f <= (float)(H_DIM - 1)) ? 1.0f : 0.0f;
            const float vy1 = (y1f >= 0.0f && y1f <= (float)(H_DIM - 1)) ? 1.0f : 0.0f;

            // fold zero-padding validity AND the z>0 mask into the weights
            const float W00 = wx0 * wy0 * vx0 * vy0 * mask;
            const float W10 = wx1 * wy0 * vx1 * vy0 * mask;
            const float W01 = wx0 * wy1 * vx0 * vy1 * mask;
            const float W11 = wx1 * wy1 * vx1 * vy1 * mask;

            const int x0i = (int)fminf(fmaxf(x0f, 0.0f), (float)(W_DIM - 1));
            const int x1i = (int)fminf(fmaxf(x1f, 0.0f), (float)(W_DIM - 1));
            const int y0i = (int)fminf(fmaxf(y0f, 0.0f), (float)(H_DIM - 1));
            const int y1i = (int)fminf(fmaxf(y1f, 0.0f), (float)(H_DIM - 1));
            const int o00 = y0i * W_DIM + x0i;
            const int o10 = y0i * W_DIM + x1i;
            const int o01 = y1i * W_DIM + x0i;
            const int o11 = y1i * W_DIM + x1i;

            const float* base = src_feats + (size_t)s * C_CH * N_PIX;
            #pragma unroll
            for (int j = 0; j < 4; ++j) {
                const int c0 = 4 * j + 2 * h;
                const float* f0 = base + c0 * N_PIX;
                const float* f1 = f0 + N_PIX;
                v2f a;
                a.x = fmaf(W00, f0[o00], fmaf(W10, f0[o10], fmaf(W01, f0[o01], W11 * f0[o11])));
                a.y = fmaf(W00, f1[o00], fmaf(W10, f1[o10], fmaf(W01, f1[o01], W11 * f1[o11])));
                // D = A(16x4,f32) x B(4x16,f32) + C  -> v_wmma_f32_16x16x4_f32
                cacc = __builtin_amdgcn_wmma_f32_16x16x4_f32(
                    false, a, false, bmat[j], (short)0, cacc, false, false);
            }
        }

        // ---- extract diagonal D[p][p]:
        //   p < 8 : lane p,      element p
        //   p >= 8: lane p + 16, element p - 8
        // Branchless binary select tree on bits of e -> 7 v_cndmask, no exec churn.
        const int e = lane & 7;
        const bool b0 = (e & 1) != 0;
        const bool b1 = (e & 2) != 0;
        const bool b2 = (e & 4) != 0;
        const float s01 = b0 ? cacc[1] : cacc[0];
        const float s23 = b0 ? cacc[3] : cacc[2];
        const float s45 = b0 ? cacc[5] : cacc[4];
        const float s67 = b0 ? cacc[7] : cacc[6];
        const float t03 = b1 ? s23 : s01;
        const float t47 = b1 ? s67 : s45;
        const float val = b2 ? t47 : t03;

        const bool writer = (lane < 8) || (lane >= 24);
        if (writer) {
            const int p = (lane < 8) ? lane : (lane - 16);   // 0..15; matches this lane's m
            out[dg * N_PIX + row * W_DIM + colb + p] = val / (msum + 1e-4f);
        }
    }
}

extern "C" void kernel_launch(void* const* d_in, const int* in_sizes, int n_in,
                              void* d_out, int out_size, void* d_ws, size_t ws_size,
                              hipStream_t stream) {
    (void)in_sizes; (void)n_in; (void)out_size; (void)d_ws; (void)ws_size;
    const float* cur_feats = (const float*)d_in[0];
    const float* src_feats = (const float*)d_in[1];
    const float* src_ext   = (const float*)d_in[2];
    // d_in[3] = src_poses (unused by reference)
    const float* src_Ks    = (const float*)d_in[4];
    const float* cur_invK  = (const float*)d_in[5];
    const float* min_depth = (const float*)d_in[6];
    const float* max_depth = (const float*)d_in[7];
    float* out = (float*)d_out;

    dim3 grid(PIX_TILES * DEPTH_TILES);   // 1536 wave32 workgroups
    dim3 block(32);
    hipLaunchKernelGGL(FeatureVolume_cost_kernel, grid, block, 0, stream,
                       cur_feats, src_feats, src_ext, src_Ks, cur_invK,
                       min_depth, max_depth, out);
}